// ShuffleFaceNet_78915729096990
// MI455X (gfx1250) — compile-verified
//
#include <hip/hip_runtime.h>
#include <cstdint>
#include <cstddef>

// ---------------------------------------------------------------------------
// fp32 WMMA fragment types (16x16x4 f32: A=16x4 -> 2 VGPR, B=4x16 -> 2 VGPR,
// C/D = 16x16 f32 -> 8 VGPR)
// ---------------------------------------------------------------------------
typedef float v2f __attribute__((ext_vector_type(2)));
typedef float v8f __attribute__((ext_vector_type(8)));

__device__ __forceinline__ float bn_fold(float x, float g, float b, float m, float v) {
    float s = g * rsqrtf(v + 1e-5f);
    return x * s + (b - m * s);
}

__device__ __forceinline__ int imin(int a, int b) { return a < b ? a : b; }

// ---------------------------------------------------------------------------
// conv1: 3x3 stride2 pad1, Cin=3 -> Cout=24, fused BN + PReLU (scalar alpha)
// grid = (spatial chunks, Cout, N) -> no div/mod by C or N
// ---------------------------------------------------------------------------
__global__ void k_conv3x3_s2_bn_prelu(const float* __restrict__ x,
                                      const float* __restrict__ w,
                                      const float* __restrict__ gg, const float* __restrict__ bb,
                                      const float* __restrict__ mm, const float* __restrict__ vv,
                                      const float* __restrict__ alpha,
                                      float* __restrict__ out,
                                      int Cin, int Cout,
                                      int Hin, int Win, int Hout, int Wout) {
    int sp = blockIdx.x * blockDim.x + threadIdx.x;
    if (sp >= Hout * Wout) return;
    int co = blockIdx.y;
    int n  = blockIdx.z;
    int wo = sp % Wout, ho = sp / Wout;
    int hi0 = ho * 2 - 1, wi0 = wo * 2 - 1;
    float acc = 0.f;
    for (int ci = 0; ci < Cin; ++ci) {
        const float* xp = x + ((size_t)(n * Cin + ci)) * Hin * Win;
        const float* wp = w + (size_t)(co * Cin + ci) * 9;
#pragma unroll
        for (int kh = 0; kh < 3; ++kh) {
            int hi = hi0 + kh;
            if ((unsigned)hi >= (unsigned)Hin) continue;
#pragma unroll
            for (int kw = 0; kw < 3; ++kw) {
                int wi = wi0 + kw;
                if ((unsigned)wi >= (unsigned)Win) continue;
                acc += xp[hi * Win + wi] * wp[kh * 3 + kw];
            }
        }
    }
    float y = bn_fold(acc, gg[co], bb[co], mm[co], vv[co]);
    float al = alpha[0];
    out[((size_t)(n * Cout + co)) * Hout * Wout + sp] = y >= 0.f ? y : al * y;
}

// ---------------------------------------------------------------------------
// maxpool 3x3 stride2 pad1  (grid = (spatial chunks, C, N))
// ---------------------------------------------------------------------------
__global__ void k_maxpool3x3_s2(const float* __restrict__ in, float* __restrict__ out,
                                int C, int Hin, int Win, int Hout, int Wout) {
    int sp = blockIdx.x * blockDim.x + threadIdx.x;
    if (sp >= Hout * Wout) return;
    int c = blockIdx.y, n = blockIdx.z;
    int wo = sp % Wout, ho = sp / Wout;
    const float* xp = in + ((size_t)(n * C + c)) * Hin * Win;
    float mx = -INFINITY;
#pragma unroll
    for (int kh = 0; kh < 3; ++kh) {
        int hi = ho * 2 - 1 + kh;
        if ((unsigned)hi >= (unsigned)Hin) continue;
#pragma unroll
        for (int kw = 0; kw < 3; ++kw) {
            int wi = wo * 2 - 1 + kw;
            if ((unsigned)wi >= (unsigned)Win) continue;
            float v = xp[hi * Win + wi];
            mx = v > mx ? v : mx;
        }
    }
    out[((size_t)(n * C + c)) * Hout * Wout + sp] = mx;
}

// ---------------------------------------------------------------------------
// generic depthwise conv (k x k, stride, pad) + BN + optional PReLU
// grid = (spatial chunks, C, N)
// ---------------------------------------------------------------------------
__global__ void k_dw_bn(const float* __restrict__ in, const float* __restrict__ w,
                        const float* __restrict__ gg, const float* __restrict__ bb,
                        const float* __restrict__ mm, const float* __restrict__ vv,
                        const float* __restrict__ alpha,
                        float* __restrict__ out,
                        int C, int Hin, int Win, int Hout, int Wout,
                        int ksz, int stride, int pad) {
    int sp = blockIdx.x * blockDim.x + threadIdx.x;
    if (sp >= Hout * Wout) return;
    int c = blockIdx.y, n = blockIdx.z;
    int wo = sp % Wout, ho = sp / Wout;
    const float* xp = in + ((size_t)(n * C + c)) * Hin * Win;
    const float* wp = w + (size_t)c * ksz * ksz;
    float acc = 0.f;
    for (int kh = 0; kh < ksz; ++kh) {
        int hi = ho * stride - pad + kh;
        if ((unsigned)hi >= (unsigned)Hin) continue;
        for (int kw = 0; kw < ksz; ++kw) {
            int wi = wo * stride - pad + kw;
            if ((unsigned)wi >= (unsigned)Win) continue;
            acc += xp[hi * Win + wi] * wp[kh * ksz + kw];
        }
    }
    float y = bn_fold(acc, gg[c], bb[c], mm[c], vv[c]);
    if (alpha) { float al = alpha[0]; y = y >= 0.f ? y : al * y; }
    out[((size_t)(n * C + c)) * Hout * Wout + sp] = y;
}

// ---------------------------------------------------------------------------
// pointwise (1x1) conv as WMMA GEMM: out[co][p] = sum_ci W[co][ci] * X[ci][p]
// fused BN + PReLU + optional channel-shuffle on store.
//
// One wave computes a 16(Cout) x 64(pixel) tile: one A fragment shared by
// 4 pixel sub-tiles / 4 accumulators -> 4 WMMAs per K-step.
// K-loop uses hand-strength-reduced running pointers (no 64-bit multiplies
// in the loop) and unconditional clamped-index loads (no EXEC divergence).
// grid = (ceil(HW/64), ceil(Cout/16), N), block = 32 (one wave).
// ---------------------------------------------------------------------------
#define PT 4
__global__ void k_pw_wmma_bn_prelu(const float* __restrict__ in, int in_ch_off, int CinT,
                                   int K, int HW,
                                   const float* __restrict__ w,
                                   const float* __restrict__ gg, const float* __restrict__ bb,
                                   const float* __restrict__ mm, const float* __restrict__ vv,
                                   const float* __restrict__ alpha,
                                   float* __restrict__ out, int CoutT, int Cout,
                                   int out_ch_off, int do_shuffle, int cpg) {
    int lane = threadIdx.x;            // 0..31 (wave32)
    int p0  = blockIdx.x * (16 * PT);
    int co0 = blockIdx.y * 16;
    int n   = blockIdx.z;
    int hf = lane >> 4;                // half-wave select
    int lm = lane & 15;
    const float* X = in + ((size_t)n * CinT + in_ch_off) * HW;

    // clamped (always in-bounds) indices
    int arowc = imin(co0 + lm, Cout - 1);          // A row (M)

    // A: 16x4 f32 — lanes0-15: M=lane,K=k..k+1; lanes16-31: M=lane-16,K=k+2..k+3
    const float* pa = w + (size_t)arowc * K + hf * 2;          // += 4 per step

    // B running pointers: pb0 -> X[kk][pix], pb1 -> X[kk+1][pix]; += 4*HW per step
    const float* pb0[PT];
    const float* pb1[PT];
#pragma unroll
    for (int t = 0; t < PT; ++t) {
        int ppc = imin(p0 + t * 16 + lm, HW - 1);
        pb0[t] = X + (size_t)(hf * 2) * HW + ppc;
        pb1[t] = pb0[t] + HW;
    }
    size_t stepB = (size_t)4 * HW;

    v8f acc[PT];
#pragma unroll
    for (int t = 0; t < PT; ++t) acc[t] = (v8f){};

    for (int k = 0; k < K; k += 4) {
        v2f a = *(const v2f*)pa;                   // 8B-aligned (K%4==0)
        pa += 4;
#pragma unroll
        for (int t = 0; t < PT; ++t) {
            v2f bfr;
            bfr.x = *pb0[t];
            bfr.y = *pb1[t];
            pb0[t] += stepB;
            pb1[t] += stepB;
            acc[t] = __builtin_amdgcn_wmma_f32_16x16x4_f32(false, a, false, bfr,
                                                           (short)0, acc[t], false, false);
        }
    }

    // ---- epilogue: BN + PReLU + (optional) channel-shuffle store ----
    float al = alpha ? alpha[0] : 0.f;
    float* outn = out + (size_t)n * CoutT * HW;
#pragma unroll
    for (int r = 0; r < 8; ++r) {
        int co = co0 + r + hf * 8;     // C/D layout: VGPR r, lanes16-31 carry M+8
        if (co >= Cout) continue;
        float s  = gg[co] * rsqrtf(vv[co] + 1e-5f);
        float sh = bb[co] - mm[co] * s;
        int gc  = out_ch_off + co;     // channel in the virtual concat tensor
        int dst = do_shuffle ? ((gc % cpg) * 2 + gc / cpg) : gc;
        float* orow = outn + (size_t)dst * HW;
#pragma unroll
        for (int t = 0; t < PT; ++t) {
            int pp = p0 + t * 16 + lm;
            if (pp >= HW) continue;
            float y = acc[t][r] * s + sh;
            if (alpha) y = y >= 0.f ? y : al * y;
            orow[pp] = y;
        }
    }
}

// ---------------------------------------------------------------------------
// copy first bf channels of `in` into `out` at shuffled positions (x1 path)
// grid = (spatial chunks, bf, N)
// ---------------------------------------------------------------------------
__global__ void k_copy_shuffle(const float* __restrict__ in, float* __restrict__ out,
                               int CinT, int CoutT, int cpg, int HW) {
    int p = blockIdx.x * blockDim.x + threadIdx.x;
    if (p >= HW) return;
    int g = blockIdx.y, n = blockIdx.z;
    int dst = (g % cpg) * 2 + g / cpg;
    out[(size_t)n * CoutT * HW + (size_t)dst * HW + p] =
        in[(size_t)n * CinT * HW + (size_t)g * HW + p];
}

// ---------------------------------------------------------------------------
// FC: out[M=64][N=128] = act[M][K=1024] @ W[N][K]^T + bias, via f32 WMMA
// M,N are multiples of 16 -> unconditional loads/stores.
// ---------------------------------------------------------------------------
__global__ void k_fc_wmma(const float* __restrict__ act, const float* __restrict__ wgt,
                          const float* __restrict__ bias, float* __restrict__ out,
                          int M, int N, int K) {
    int lane = threadIdx.x;
    int n0 = blockIdx.x * 16;
    int m0 = blockIdx.y * 16;
    int hf = lane >> 4, lm = lane & 15;
    const float* pa = act + (size_t)(m0 + lm) * K + hf * 2;
    const float* pb = wgt + (size_t)(n0 + lm) * K + hf * 2;   // B[k][n] = W[n][k]
    v8f acc = {};
    for (int k = 0; k < K; k += 4) {
        v2f a   = *(const v2f*)pa;
        v2f bfr = *(const v2f*)pb;
        pa += 4; pb += 4;
        acc = __builtin_amdgcn_wmma_f32_16x16x4_f32(false, a, false, bfr,
                                                    (short)0, acc, false, false);
    }
    int nn = n0 + lm;
    float bv = bias[nn];
#pragma unroll
    for (int r = 0; r < 8; ++r) {
        int mm_ = m0 + r + hf * 8;
        out[(size_t)mm_ * N + nn] = acc[r] + bv;
    }
}

// ===========================================================================
// Host orchestration
// ===========================================================================
struct BNp { const float *g, *b, *m, *v; };

static inline void launch_pw(hipStream_t s, const float* in, int in_off, int CinT,
                             int K, int HW, const float* w, BNp bn, const float* alpha,
                             float* out, int CoutT, int Cout, int out_off,
                             int shuf, int cpg, int N) {
    dim3 grid((HW + 16 * PT - 1) / (16 * PT), (Cout + 15) / 16, N);
    k_pw_wmma_bn_prelu<<<grid, 32, 0, s>>>(in, in_off, CinT, K, HW, w,
                                           bn.g, bn.b, bn.m, bn.v, alpha,
                                           out, CoutT, Cout, out_off, shuf, cpg);
}

static inline void launch_dw(hipStream_t s, const float* in, const float* w, BNp bn,
                             const float* alpha, float* out, int N, int C,
                             int Hin, int Win, int Hout, int Wout,
                             int k, int stride, int pad) {
    dim3 grid((Hout * Wout + 255) / 256, C, N);
    k_dw_bn<<<grid, 256, 0, s>>>(in, w, bn.g, bn.b, bn.m, bn.v, alpha,
                                 out, C, Hin, Win, Hout, Wout, k, stride, pad);
}

extern "C" void kernel_launch(void* const* d_in, const int* in_sizes, int n_in,
                              void* d_out, int out_size, void* d_ws, size_t ws_size,
                              hipStream_t stream) {
    (void)in_sizes; (void)n_in; (void)out_size; (void)ws_size;

    // ---- walk d_in in setup_inputs() dict insertion order (depth-first) ----
    int ip = 0;
    auto nf = [&]() -> const float* { return (const float*)d_in[ip++]; };

    const float* x = nf();
    const float* conv1_w = nf();
    BNp bn1 = { nf(), nf(), nf(), nf() };
    const float* a1 = nf();

    struct Blk {
        BNp bn2; const float* a; const float* dw2; const float* c3;
        const float* dw1; BNp bnb1; const float* c1; const float* c2; int s2;
    };
    const int REP[3] = {4, 8, 4};
    const int OUP[3] = {48, 96, 192};
    Blk blocks[16]; int nb = 0;
    for (int si = 0; si < 3; ++si) {
        for (int bi = 0; bi < REP[si]; ++bi) {
            Blk bk{};
            bk.bn2 = { nf(), nf(), nf(), nf() };
            bk.a = nf(); bk.dw2 = nf(); bk.c3 = nf();
            if (bi == 0) {
                bk.dw1 = nf(); bk.bnb1 = { nf(), nf(), nf(), nf() };
                bk.c1 = nf(); bk.c2 = nf(); bk.s2 = 1;
            } else {
                bk.c2 = nf(); bk.s2 = 0;
            }
            blocks[nb++] = bk;
        }
    }
    const float* conv5_w = nf(); BNp bn5 = { nf(), nf(), nf(), nf() }; const float* a5 = nf();
    const float* gdc1_w = nf();  BNp gbn1 = { nf(), nf(), nf(), nf() }; const float* ga1 = nf();
    const float* gdc2_w = nf();  BNp gbn2 = { nf(), nf(), nf(), nf() }; const float* ga2 = nf();
    const float* fc_w = nf();    const float* fc_b = nf();

    // ---- workspace layout ----
    char* ws = (char*)d_ws;
    float* bufA = (float*)ws;                          // 80 MB: conv1 out + ping
    float* bufB = (float*)(ws + (80ull  << 20));       // 24 MB: pong
    float* t1   = (float*)(ws + (104ull << 20));       // 24 MB: block temp 1
    float* t2   = (float*)(ws + (128ull << 20));       // 24 MB: block temp 2

    const int N = 64;

    // conv1 + bn + prelu: (64,3,224,224) -> (64,24,112,112)
    {
        dim3 grid((112 * 112 + 255) / 256, 24, N);
        k_conv3x3_s2_bn_prelu<<<grid, 256, 0, stream>>>(
            x, conv1_w, bn1.g, bn1.b, bn1.m, bn1.v, a1, bufA,
            3, 24, 224, 224, 112, 112);
    }
    // maxpool: -> (64,24,56,56)
    {
        dim3 grid((56 * 56 + 255) / 256, 24, N);
        k_maxpool3x3_s2<<<grid, 256, 0, stream>>>(bufA, bufB, 24, 112, 112, 56, 56);
    }

    float* cur = bufB; float* nxt = bufA;
    int C = 24, H = 56, W = 56, bidx = 0;

    for (int si = 0; si < 3; ++si) {
        int oup = OUP[si], bf = oup / 2;
        for (int bi = 0; bi < REP[si]; ++bi) {
            Blk& bk = blocks[bidx++];
            if (bk.s2) {
                int Ho = H / 2, Wo = W / 2;
                int HWin = H * W, HWout = Ho * Wo;
                // branch1: dw3x3 s2 (+bn1) then 1x1 c1 (+bn2+prelu) -> shuffled ch [0,bf)
                launch_dw(stream, cur, bk.dw1, bk.bnb1, nullptr, t1,
                          N, C, H, W, Ho, Wo, 3, 2, 1);
                launch_pw(stream, t1, 0, C, C, HWout, bk.c1, bk.bn2, bk.a,
                          nxt, oup, bf, 0, 1, bf, N);
                // branch2: 1x1 c2 -> dw3x3 s2 -> 1x1 c3 -> shuffled ch [bf,2bf)
                launch_pw(stream, cur, 0, C, C, HWin, bk.c2, bk.bn2, bk.a,
                          t1, bf, bf, 0, 0, bf, N);
                launch_dw(stream, t1, bk.dw2, bk.bn2, nullptr, t2,
                          N, bf, H, W, Ho, Wo, 3, 2, 1);
                launch_pw(stream, t2, 0, bf, bf, HWout, bk.c3, bk.bn2, bk.a,
                          nxt, oup, bf, bf, 1, bf, N);
                C = oup; H = Ho; W = Wo;
            } else {
                int HW = H * W;   // C == oup here
                // x1 half -> shuffled positions
                {
                    dim3 grid((HW + 255) / 256, bf, N);
                    k_copy_shuffle<<<grid, 256, 0, stream>>>(cur, nxt, C, oup, bf, HW);
                }
                // branch2 on x2 half
                launch_pw(stream, cur, C / 2, C, bf, HW, bk.c2, bk.bn2, bk.a,
                          t1, bf, bf, 0, 0, bf, N);
                launch_dw(stream, t1, bk.dw2, bk.bn2, nullptr, t2,
                          N, bf, H, W, H, W, 3, 1, 1);
                launch_pw(stream, t2, 0, bf, bf, HW, bk.c3, bk.bn2, bk.a,
                          nxt, oup, bf, bf, 1, bf, N);
            }
            float* tmp = cur; cur = nxt; nxt = tmp;
        }
    }

    // conv5: 1x1 192->1024 + bn5 + prelu (HW = 49)
    launch_pw(stream, cur, 0, 192, 192, 49, conv5_w, bn5, a5,
              nxt, 1024, 1024, 0, 0, 512, N);
    { float* tmp = cur; cur = nxt; nxt = tmp; }

    // gdc1: depthwise 5x5 pad0: 7x7 -> 3x3, + bn + prelu
    launch_dw(stream, cur, gdc1_w, gbn1, ga1, nxt, N, 1024, 7, 7, 3, 3, 5, 1, 0);
    { float* tmp = cur; cur = nxt; nxt = tmp; }

    // gdc2: depthwise 3x3 pad0: 3x3 -> 1x1, + bn + prelu  -> (64,1024)
    launch_dw(stream, cur, gdc2_w, gbn2, ga2, nxt, N, 1024, 3, 3, 1, 1, 3, 1, 0);
    { float* tmp = cur; cur = nxt; nxt = tmp; }

    // FC: (64,1024) @ (128,1024)^T + b -> d_out (64,128)
    {
        dim3 grid(128 / 16, 64 / 16);
        k_fc_wmma<<<grid, 32, 0, stream>>>(cur, fc_w, fc_b, (float*)d_out, 64, 128, 1024);
    }
}